// ImproveFFBlock4_88167088652743
// MI455X (gfx1250) — compile-verified
//
#include <hip/hip_runtime.h>
#include <stdint.h>

// ---------------------------------------------------------------------------
// Problem shapes (hardcoded from the reference):
//   in1: [16,128, 48, 48]  -> pool2  -> p1 [16,128,24,24]
//   in2: [16, 64, 96, 96]  -> pool4  -> p2 [16, 64,24,24]
//   in3: [16, 32,192,192]  -> pool8  -> p3 [16, 32,24,24]
//   in4: [16, 16,384,384]  -> pool16 -> p4 [16, 16,24,24]
//   out[b,c,h,w] = relu(p1[b,c%128,h,w] + p2[b,c%64,h,w] + p3[b,c%32,h,w]
//                       + p4[b,c%16,h,w] + ff[b,c,h,w])   [16,256,24,24]
// Memory-bound: ~302 MB unique traffic => ~13 us at 23.3 TB/s. TDM streams
// the pooling windows into LDS; no matrix ops exist in this workload.
// ---------------------------------------------------------------------------

typedef uint32_t u32x4 __attribute__((ext_vector_type(4)));
typedef uint32_t u32x8 __attribute__((ext_vector_type(8)));

// Low 32 bits of a generic pointer to __shared__ data are the LDS byte offset
// (shared aperture lives entirely in addr[63:32] per CDNA5 ISA §10.2).
__device__ __forceinline__ uint32_t lds_byte_addr(const void* p) {
  return (uint32_t)(uintptr_t)p;
}

// Issue a TDM load of a 2D tile (rows x width f32 elements, row stride in
// elements) from global memory into LDS. D# descriptor built per
// CDNA5 ISA; issued as inline asm (portable across the
// two toolchains' differing builtin arities). Tracked by TENSORcnt.
// Must be executed by exactly one wave per transfer.
__device__ __forceinline__ void tdm_load_rows_to_lds(uint32_t lds_addr,
                                                     const float* gsrc,
                                                     uint32_t width_elems,
                                                     uint32_t rows,
                                                     uint32_t row_stride_elems) {
  uint64_t ga = (uint64_t)(uintptr_t)gsrc;
  u32x4 g0;
  g0[0] = 1u;                                              // count=1, user D#
  g0[1] = lds_addr;                                        // lds_addr [63:32]
  g0[2] = (uint32_t)ga;                                    // global_addr lo
  g0[3] = ((uint32_t)(ga >> 32) & 0x01FFFFFFu)             // global_addr [56:32]
          | (2u << 30);                                    // type=2 ("image")
  u32x8 g1;
  g1[0] = 2u << 16;                                        // data_size=2 (4B)
  g1[1] = (width_elems & 0xFFFFu) << 16;                   // tensor_dim0 lo16
  g1[2] = ((width_elems >> 16) & 0xFFFFu)                  // tensor_dim0 hi16
          | ((rows & 0xFFFFu) << 16);                      // tensor_dim1 lo16
  g1[3] = ((rows >> 16) & 0xFFFFu)                         // tensor_dim1 hi16
          | ((width_elems & 0xFFFFu) << 16);               // tile_dim0
  g1[4] = rows & 0xFFFFu;                                  // tile_dim1 (tile_dim2=0)
  g1[5] = row_stride_elems;                                // tensor_dim0_stride lo32
  g1[6] = 0u;                                              // stride hi / dim1_stride
  g1[7] = 0u;
  u32x4 g2 = {0u, 0u, 0u, 0u};                             // dims 2..3 unused
  u32x4 g3 = {0u, 0u, 0u, 0u};                             // dim 4 unused
  asm volatile("tensor_load_to_lds %0, %1, %2, %3"
               :
               : "s"(g0), "s"(g1), "s"(g2), "s"(g3)
               : "memory");
}

// ---------------------------------------------------------------------------
// Pass 1a: 2x2 max pool of [2048 planes, 48x48] -> [2048 planes, 24x24].
// Plain float4 streaming; each thread produces 2 adjacent outputs.
// ---------------------------------------------------------------------------
__global__ void pool2_kernel(const float4* __restrict__ in4,
                             float2* __restrict__ out2) {
  int tid = blockIdx.x * blockDim.x + threadIdx.x;   // 0 .. 589,823
  int plane = tid / 288;
  int r = tid - plane * 288;
  int h = r / 12;
  int q = r - h * 12;                       // float4 column index (0..11)
  size_t base = (size_t)plane * 576 + (size_t)(2 * h) * 12 + q;
  float4 a = in4[base];
  float4 b = in4[base + 12];
  float m0 = fmaxf(fmaxf(a.x, a.y), fmaxf(b.x, b.y));
  float m1 = fmaxf(fmaxf(a.z, a.w), fmaxf(b.z, b.w));
  out2[(size_t)plane * 288 + h * 12 + q] = make_float2(m0, m1);
}

// ---------------------------------------------------------------------------
// Pass 1b: KxK max pool via TDM double-buffered row-group staging.
//   W   = 24*K input width (== blockDim.x)
//   RPB = output rows handled per block (grid = planes * (24/RPB))
//   R   = output rows staged per TDM transfer (tile = R*K rows x W floats)
// Wave 0 owns the TDM pipeline; the block does a conflict-free column max
// from LDS then a wave32 shfl_xor horizontal max per K-lane group.
// ---------------------------------------------------------------------------
template <int K, int W, int RPB, int R>
__global__ void __launch_bounds__(W) pool_tdm_kernel(const float* __restrict__ in,
                                                     float* __restrict__ out) {
  constexpr int ITERS = RPB / R;
  constexpr int BPB = 24 / RPB;             // blocks per plane (power of 2)
  const int plane = blockIdx.x / BPB;
  const int part  = blockIdx.x - plane * BPB;
  const int oh0   = part * RPB;             // first output row of this block
  const float* pin = in + (size_t)plane * (size_t)(24 * K) * W;
  float* pout = out + (size_t)plane * 576;
  __shared__ float buf[2][R * K * W];
  const int t = threadIdx.x;
  const bool issuer = (t < 32);             // wave 0 issues all TDM ops

  if (issuer) {
    tdm_load_rows_to_lds(lds_byte_addr(&buf[0][0]),
                         pin + (size_t)oh0 * K * W,
                         (uint32_t)W, (uint32_t)(R * K), (uint32_t)W);
  }
  for (int i = 0; i < ITERS; ++i) {
    if (issuer) {
      if (i + 1 < ITERS) {
        tdm_load_rows_to_lds(lds_byte_addr(&buf[(i + 1) & 1][0]),
                             pin + (size_t)(oh0 + (i + 1) * R) * K * W,
                             (uint32_t)W, (uint32_t)(R * K), (uint32_t)W);
        __builtin_amdgcn_s_wait_tensorcnt(1);   // transfer(i) done (in-order)
      } else {
        __builtin_amdgcn_s_wait_tensorcnt(0);   // drain on last iteration
      }
    }
    __syncthreads();                            // buf[i&1] visible to all waves
    const float* b = buf[i & 1];
#pragma unroll
    for (int r = 0; r < R; ++r) {
      float m = b[r * K * W + t];
#pragma unroll
      for (int rr = 1; rr < K; ++rr) m = fmaxf(m, b[(r * K + rr) * W + t]);
#pragma unroll
      for (int s = 1; s < K; s <<= 1) m = fmaxf(m, __shfl_xor(m, s));
      if ((t & (K - 1)) == 0) pout[(size_t)(oh0 + i * R + r) * 24 + (t / K)] = m;
    }
    __syncthreads();   // readers done before transfer(i+2) overwrites buf[i&1]
  }
}

// ---------------------------------------------------------------------------
// Pass 2: gather the 4 pooled planes (channel-tiled) + pure_ff, add, ReLU.
// Vectorized float4 over the 576-element spatial plane (144 float4s).
// Pooled intermediates (8.9 MB) hit in the 192 MB L2.
// ---------------------------------------------------------------------------
__global__ void combine_kernel(const float4* __restrict__ p1,
                               const float4* __restrict__ p2,
                               const float4* __restrict__ p3,
                               const float4* __restrict__ p4,
                               const float4* __restrict__ ff,
                               float4* __restrict__ out) {
  int i = blockIdx.x * blockDim.x + threadIdx.x;   // 0 .. 589,823
  int s = i % 144;
  int bc = i / 144;
  int c = bc & 255;
  int b = bc >> 8;
  float4 v1 = p1[(size_t)(b * 128 + (c & 127)) * 144 + s];
  float4 v2 = p2[(size_t)(b * 64 + (c & 63)) * 144 + s];
  float4 v3 = p3[(size_t)(b * 32 + (c & 31)) * 144 + s];
  float4 v4 = p4[(size_t)(b * 16 + (c & 15)) * 144 + s];
  float4 f = ff[i];
  float4 r;
  r.x = fmaxf(v1.x + v2.x + v3.x + v4.x + f.x, 0.0f);
  r.y = fmaxf(v1.y + v2.y + v3.y + v4.y + f.y, 0.0f);
  r.z = fmaxf(v1.z + v2.z + v3.z + v4.z + f.z, 0.0f);
  r.w = fmaxf(v1.w + v2.w + v3.w + v4.w + f.w, 0.0f);
  out[i] = r;
}

extern "C" void kernel_launch(void* const* d_in, const int* in_sizes, int n_in,
                              void* d_out, int out_size, void* d_ws, size_t ws_size,
                              hipStream_t stream) {
  const float* in1 = (const float*)d_in[0];   // [16,128,48,48]
  const float* in2 = (const float*)d_in[1];   // [16,64,96,96]
  const float* in3 = (const float*)d_in[2];   // [16,32,192,192]
  const float* in4 = (const float*)d_in[3];   // [16,16,384,384]
  const float* ff  = (const float*)d_in[4];   // [16,256,24,24]

  // Workspace layout (floats): p1 | p2 | p3 | p4  = 8.85 MB total.
  float* p1 = (float*)d_ws;                   // 16*128*576
  float* p2 = p1 + (size_t)16 * 128 * 576;    // 16*64*576
  float* p3 = p2 + (size_t)16 * 64 * 576;     // 16*32*576
  float* p4 = p3 + (size_t)16 * 32 * 576;     // 16*16*576

  // Pass 1: pooled planes (each input read exactly once).
  pool2_kernel<<<2304, 256, 0, stream>>>((const float4*)in1, (float2*)p1);
  // K=4:  1024 planes * 2 blocks, 9 KB per TDM transfer (24 rows x 96).
  pool_tdm_kernel<4, 96, 12, 6><<<16 * 64 * 2, 96, 0, stream>>>(in2, p2);
  // K=8:  512 planes * 2 blocks, 18 KB per TDM transfer (24 rows x 192).
  pool_tdm_kernel<8, 192, 12, 3><<<16 * 32 * 2, 192, 0, stream>>>(in3, p3);
  // K=16: 256 planes * 4 blocks, 24 KB per TDM transfer (16 rows x 384).
  pool_tdm_kernel<16, 384, 6, 1><<<16 * 16 * 4, 384, 0, stream>>>(in4, p4);

  // Pass 2: gather + add + ReLU.
  combine_kernel<<<2304, 256, 0, stream>>>(
      (const float4*)p1, (const float4*)p2, (const float4*)p3,
      (const float4*)p4, (const float4*)ff, (float4*)d_out);
}